// RasterizePointsXYsBlending_9337258902371
// MI455X (gfx1250) — compile-verified
//
#include <hip/hip_runtime.h>

typedef float v2f __attribute__((ext_vector_type(2)));
typedef float v8f __attribute__((ext_vector_type(8)));

namespace {
constexpr int B_ = 2;
constexpr int N_ = 2048;
constexpr int C_ = 64;
constexpr int S_ = 128;
constexpr int K_ = 8;
constexpr int TS = 16;              // pixel tile edge
constexpr int TG = S_ / TS;         // 8 tiles per dim
constexpr int NT = TG * TG;         // 64 tiles per image
constexpr float RADIUS_NDC = 1.5f / S_ * 2.0f;          // 0.0234375
constexpr float R2 = RADIUS_NDC * RADIUS_NDC;
constexpr float RPIX = 1.5f;        // radius in pixel units = R*S/2
}

__global__ void init_cnt_kernel(int* __restrict__ cnt) {
    if (threadIdx.x < B_ * NT) cnt[threadIdx.x] = 0;
}

// One thread per (b, point): append (xe, ye, z, idx) to every tile whose
// pixel-center range the point's radius can touch (conservative; exact test later).
__global__ void bin_kernel(const float* __restrict__ pts, int* __restrict__ cnt,
                           float4* __restrict__ lists, int cap) {
    int gid = blockIdx.x * 256 + threadIdx.x;
    if (gid >= B_ * N_) return;
    int b = gid / N_;
    int n = gid - b * N_;
    const float* p = pts + (size_t)gid * 3;
    float xe = -p[0], ye = -p[1], z = p[2];
    if (!(z > 0.0f)) return;
    // pixel-space center: px = ((1 - v)*S - 1)/2 ; candidates within RPIX (+slack)
    float pxc = ((1.0f - xe) * (float)S_ - 1.0f) * 0.5f;
    float pyc = ((1.0f - ye) * (float)S_ - 1.0f) * 0.5f;
    int ix0 = (int)floorf(pxc - RPIX);
    int ix1 = (int)ceilf (pxc + RPIX);
    int iy0 = (int)floorf(pyc - RPIX);
    int iy1 = (int)ceilf (pyc + RPIX);
    if (ix0 < 0) ix0 = 0;
    if (iy0 < 0) iy0 = 0;
    if (ix1 > S_ - 1) ix1 = S_ - 1;
    if (iy1 > S_ - 1) iy1 = S_ - 1;
    if (ix0 > ix1 || iy0 > iy1) return;
    float4 rec = make_float4(xe, ye, z, __int_as_float(n));
    for (int tyT = iy0 >> 4; tyT <= (iy1 >> 4); ++tyT) {
        for (int txT = ix0 >> 4; txT <= (ix1 >> 4); ++txT) {
            int tid = b * NT + tyT * TG + txT;
            int slot = atomicAdd(&cnt[tid], 1);
            if (slot < cap) lists[(size_t)tid * cap + slot] = rec;
        }
    }
}

// One 256-thread block per (tile, b): per-pixel K=8 selection, then per-wave
// WMMA f32 16x16x4 compositing of 16-pixel rows against gathered features.
__global__ __launch_bounds__(256)
void composite_kernel(const float* __restrict__ src, const int* __restrict__ cnt,
                      const float4* __restrict__ lists, int cap,
                      float* __restrict__ out) {
    __shared__ float4 cbuf[256];
    __shared__ float  w_s[256][K_];
    __shared__ int    id_s[256][K_];

    const int tile = blockIdx.x;
    const int b    = blockIdx.y;
    const int tY = tile >> 3, tX = tile & 7;
    const int t  = threadIdx.x;
    const int tx = t & 15, ty = t >> 4;
    const float pixX = 1.0f - (2.0f * (float)(tX * TS + tx) + 1.0f) / (float)S_;
    const float pixY = 1.0f - (2.0f * (float)(tY * TS + ty) + 1.0f) / (float)S_;

    int nv = cnt[b * NT + tile];
    if (nv > cap) nv = cap;
    const float4* list = lists + (size_t)(b * NT + tile) * cap;

    const float INF = __int_as_float(0x7f800000);
    float kz[K_], kd[K_];
    int   kid[K_];
#pragma unroll
    for (int k = 0; k < K_; ++k) { kz[k] = INF; kd[k] = 0.0f; kid[k] = 0; }

    for (int base = 0; base < nv; base += 256) {
        int m = nv - base;
        if (m > 256) m = 256;
        __syncthreads();
        if (t < m) cbuf[t] = list[base + t];
        __syncthreads();
        for (int j = 0; j < m; ++j) {
            float4 pr = cbuf[j];
            float dx = pixX - pr.x;
            float dy = pixY - pr.y;
            float d2 = dy * dy + dx * dx;
            if (d2 < R2) {
                float z = pr.z;
                int   n = __float_as_int(pr.w);
                if (z < kz[K_ - 1] || (z == kz[K_ - 1] && n < kid[K_ - 1])) {
                    kz[K_ - 1] = z; kd[K_ - 1] = d2; kid[K_ - 1] = n;
#pragma unroll
                    for (int j2 = K_ - 1; j2 > 0; --j2) {
                        bool sw = (kz[j2] < kz[j2 - 1]) ||
                                  (kz[j2] == kz[j2 - 1] && kid[j2] < kid[j2 - 1]);
                        if (sw) {
                            float tz = kz[j2]; kz[j2] = kz[j2 - 1]; kz[j2 - 1] = tz;
                            float td = kd[j2]; kd[j2] = kd[j2 - 1]; kd[j2 - 1] = td;
                            int   ti = kid[j2]; kid[j2] = kid[j2 - 1]; kid[j2 - 1] = ti;
                        }
                    }
                }
            }
        }
    }

    // alphas -> front-to-back weights (TAU=1, RAD_POW=2)
    {
        float T = 1.0f;
#pragma unroll
        for (int k = 0; k < K_; ++k) {
            float a = 0.0f;
            if (kz[k] < INF) {
                float dn = kd[k] * (1.0f / R2);
                dn = fminf(fmaxf(dn, 0.001f), 1.0f);
                a = 1.0f - sqrtf(dn);     // > 0 for every valid slot
            }
            w_s[t][k]  = a * T;           // == 0 iff slot invalid (prefix-valid)
            id_s[t][k] = kid[k];
            T *= (1.0f - a);
        }
    }
    __syncthreads();

    // WMMA compositing: per wave, two 16-pixel rows; out(16x64) = W(16x128)*G(128x64)
    const int lane = t & 31;
    const int wv   = t >> 5;
    const int half = lane >> 4;     // 0: K0/K1 lanes, 1: K2/K3 lanes
    const int lm   = lane & 15;
    const float* srcb = src + (size_t)b * C_ * N_;
    float* outb = out + (size_t)b * C_ * S_ * S_;

    for (int g = 0; g < 2; ++g) {
        v8f acc[4];
#pragma unroll
        for (int ct = 0; ct < 4; ++ct) {
            v8f zer = {0.f, 0.f, 0.f, 0.f, 0.f, 0.f, 0.f, 0.f};
            acc[ct] = zer;
        }
        const int Y = tY * TS + wv * 2 + g;
        for (int s = 0; s < 32; ++s) {
            const int p0 = s >> 1;              // pixel (row M) owning this K-slice
            const int kb = (s & 1) * 4;         // slot base within the pixel
            const int pslot = wv * 32 + g * 16 + p0;
            if (w_s[pslot][kb] == 0.0f) continue;   // uniform: weights prefix-valid
            const int sl = kb + half * 2;
            float wa = w_s[pslot][sl + 0];
            float wb = w_s[pslot][sl + 1];
            int   i0 = id_s[pslot][sl + 0];
            int   i1 = id_s[pslot][sl + 1];
            bool mine = (lm == p0);
            v2f A;
            A.x = mine ? wa : 0.0f;             // A: 16x4, row p0 nonzero only
            A.y = mine ? wb : 0.0f;
#pragma unroll
            for (int ct = 0; ct < 4; ++ct) {
                const float* row = srcb + (size_t)(ct * 16 + lm) * N_;
                v2f Bv;
                Bv.x = row[i0];                 // B: 4x16 gathered feature slice
                Bv.y = row[i1];
                acc[ct] = __builtin_amdgcn_wmma_f32_16x16x4_f32(
                    false, A, false, Bv, (short)0, acc[ct], false, false);
            }
        }
        // C/D layout: VGPR rr -> M = rr + half*8 ; N = lane&15
#pragma unroll
        for (int ct = 0; ct < 4; ++ct) {
            const int c = ct * 16 + lm;
#pragma unroll
            for (int rr = 0; rr < 8; ++rr) {
                const int X = tX * TS + half * 8 + rr;
                outb[((size_t)c * S_ + Y) * S_ + X] = acc[ct][rr];
            }
        }
    }
}

extern "C" void kernel_launch(void* const* d_in, const int* in_sizes, int n_in,
                              void* d_out, int out_size, void* d_ws, size_t ws_size,
                              hipStream_t stream) {
    (void)in_sizes; (void)n_in; (void)out_size;
    const float* pts = (const float*)d_in[0];   // [B,N,3]
    const float* src = (const float*)d_in[1];   // [B,C,N]
    float* out = (float*)d_out;                 // [B,C,S,S]

    int* cnt = (int*)d_ws;                                  // B*NT counters
    float4* lists = (float4*)((char*)d_ws + 1024);
    size_t per_slot = (size_t)B_ * NT * sizeof(float4);
    size_t cap_avail = ws_size > 1024 ? (ws_size - 1024) / per_slot : 0;
    int cap = (int)(cap_avail < (size_t)N_ ? cap_avail : (size_t)N_);
    if (cap < 1) cap = 1;

    init_cnt_kernel<<<1, 128, 0, stream>>>(cnt);
    bin_kernel<<<(B_ * N_ + 255) / 256, 256, 0, stream>>>(pts, cnt, lists, cap);
    composite_kernel<<<dim3(NT, B_), 256, 0, stream>>>(src, cnt, lists, cap, out);
}